// Spherization_90374701842707
// MI455X (gfx1250) — compile-verified
//
#include <hip/hip_runtime.h>
#include <math.h>

// Spherization, MI455X (gfx1250, wave32).
//
// Per row (N=512): ang = (PI-2*phiL)*sigmoid(s*x)+phiL; out[j<512] =
// sign(cos(ang_j)) * r * exp( excl_prefix_j(log|sin|+eps) + log(|cos_j|+eps) );
// out[512] = r * exp( full_sum(log|sin|+eps) ).
//
// One wave handles 16 rows. Columns processed in 32 chunks of 16.
// The 16-wide exclusive scan is done as D = A x U_strict + C using four
// chained V_WMMA_F32_16X16X4_F32 (K=16). Carry per row is (D + log_sin) at
// column 15, broadcast to the next chunk's C matrix via shfl within 16-lane
// halves (matches the 16x16 f32 C/D VGPR layout: lanes0-15 rows 0-7,
// lanes16-31 rows 8-15).

typedef __attribute__((ext_vector_type(2))) float v2f;
typedef __attribute__((ext_vector_type(8))) float v8f;

#define NF 512
#define NOUT (NF + 1)
#define NCHUNK (NF / 16)
#define WAVES_PER_BLOCK 8
#define LDS_STRIDE 17   // pad: conflict-free A-layout ds_load_b64

__global__ __launch_bounds__(WAVES_PER_BLOCK * 32)
void spherization_kernel(const float* __restrict__ x,
                         const float* __restrict__ scaling,
                         const float* __restrict__ radius,
                         float* __restrict__ out,
                         float ang_scale, float phi_l, int rows)
{
    __shared__ float lstile[WAVES_PER_BLOCK][16 * LDS_STRIDE];

    const int lane = threadIdx.x & 31;
    const int wave = threadIdx.x >> 5;
    const long row_base = ((long)blockIdx.x * WAVES_PER_BLOCK + wave) * 16;
    if (row_base >= rows) return;   // wave-uniform exit (rows % 16 == 0)

    const float s   = scaling[0];
    const float rad = radius[0];

    const int half = lane >> 4;     // 0: lanes 0-15, 1: lanes 16-31
    const int nlo  = lane & 15;     // D-layout column / A-layout row

    // Strict-upper-triangular B in B-layout, split into four 4x16 K-slices.
    // Per slice g: VGPR0 holds k = 4g + 2*half, VGPR1 holds k+1, column = nlo.
    v2f bmat[4];
    #pragma unroll
    for (int g = 0; g < 4; ++g) {
        const int k0 = 4 * g + 2 * half;
        bmat[g].x = (k0     < nlo) ? 1.0f : 0.0f;
        bmat[g].y = (k0 + 1 < nlo) ? 1.0f : 0.0f;
    }

    float* myTile = lstile[wave];   // private per wave: no barriers needed

    float carry[8];                 // running per-row prefix (broadcast in half)
    #pragma unroll
    for (int r = 0; r < 8; ++r) carry[r] = 0.0f;

    for (int cchunk = 0; cchunk < NCHUNK; ++cchunk) {
        const int col = cchunk * 16 + nlo;

        float ls[8], lc[8], cs[8];
        #pragma unroll
        for (int r = 0; r < 8; ++r) {
            const long row = row_base + r + 8 * half;   // D-layout row for VGPR r
            const float xv = x[row * NF + col];
            // sigmoid -> angularize
            const float e   = __expf(-s * xv);
            const float sg  = __builtin_amdgcn_rcpf(1.0f + e);
            const float ang = ang_scale * sg + phi_l;
            const float sv  = __sinf(ang);
            const float cv  = __cosf(ang);
            ls[r] = __logf(fabsf(sv) + 1e-6f);
            lc[r] = __logf(fabsf(cv) + 1e-6f);
            cs[r] = cv;
            // stage log_sin row-major into LDS for the A-layout reload
            myTile[(r + 8 * half) * LDS_STRIDE + nlo] = ls[r];
        }
        // same-wave cross-lane LDS dependence: order + complete the stores
        asm volatile("s_wait_dscnt 0" ::: "memory");

        // A-layout reload: lane holds row m=nlo, k = 4g + 2*half (+1) -> b64
        v2f amat[4];
        #pragma unroll
        for (int g = 0; g < 4; ++g) {
            const float* p = &myTile[nlo * LDS_STRIDE + 4 * g + 2 * half];
            amat[g] = *(const v2f*)p;               // ds_load_b64, conflict-free
        }

        // Exclusive 16-wide scan with carry: D = A * U_strict + C
        v8f acc;
        #pragma unroll
        for (int r = 0; r < 8; ++r) acc[r] = carry[r];
        #pragma unroll
        for (int g = 0; g < 4; ++g) {
            acc = __builtin_amdgcn_wmma_f32_16x16x4_f32(
                false, amat[g], false, bmat[g], (short)0, acc, false, false);
        }

        // emit outputs, update carry (column 15 inclusive sum, broadcast)
        #pragma unroll
        for (int r = 0; r < 8; ++r) {
            const long row = row_base + r + 8 * half;
            const float v = rad * __expf(acc[r] + lc[r]);
            out[row * NOUT + col] = (cs[r] < 0.0f) ? -v : v;
            const float incl = acc[r] + ls[r];
            carry[r] = __shfl(incl, 15, 16);   // lane15 -> 0-15, lane31 -> 16-31
        }
    }

    // column 512: radius * exp(full row sum); sign forced positive
    if (nlo == 0) {
        #pragma unroll
        for (int r = 0; r < 8; ++r) {
            const long row = row_base + r + 8 * half;
            out[row * NOUT + NF] = rad * __expf(carry[r]);
        }
    }
}

extern "C" void kernel_launch(void* const* d_in, const int* in_sizes, int n_in,
                              void* d_out, int out_size, void* d_ws, size_t ws_size,
                              hipStream_t stream) {
    const float* x       = (const float*)d_in[0];
    const float* scaling = (const float*)d_in[1];
    const float* radius  = (const float*)d_in[2];
    float* out           = (float*)d_out;

    const int rows = in_sizes[0] / NF;       // 65536

    // constants exactly as the reference computes them (PI = 3.141592)
    const double PI_REF = 3.141592;
    double phi_l = asin(pow(1e-6, 1.0 / (double)NF));
    const double alt = PI_REF / 2.0 * (1.0 - 0.01);
    if (alt < phi_l) phi_l = alt;
    const float ang_scale = (float)(PI_REF - 2.0 * phi_l);

    const int waves  = rows / 16;
    const int blocks = (waves + WAVES_PER_BLOCK - 1) / WAVES_PER_BLOCK;
    spherization_kernel<<<blocks, WAVES_PER_BLOCK * 32, 0, stream>>>(
        x, scaling, radius, out, ang_scale, (float)phi_l, rows);
}